// DCNet_89850715832913
// MI455X (gfx1250) — compile-verified
//
#include <hip/hip_runtime.h>
#include <hip/hip_bf16.h>
#include <math.h>

// ---------------------------------------------------------------------------
// DCN (Deep & Cross Network) forward on MI455X (gfx1250, wave32).
// Dominant cost: 3 fp32 GEMMs (42.4 GFLOP) -> V_WMMA_F32_16X16X4_F32,
// weight panels streamed into LDS by the Tensor Data Mover (TDM),
// double-buffered, synchronized with s_wait_tensorcnt.
// All GEMM trip counts are compile-time (templated N/K) so the unrolled
// WMMA body has no runtime guards and no register spills.
// ---------------------------------------------------------------------------

#define B_SZ      16384
#define N_FIELDS  39
#define K_EMB     16
#define XDIM      624          // 39*16
#define FC1       1024
#define FC2       512
#define FC3       256
#define ZDIM      880          // FC3 + XDIM
#define NNZ       (B_SZ * N_FIELDS)
#define KC        64           // K-chunk staged in LDS per block iteration

typedef __attribute__((ext_vector_type(2))) float v2f;
typedef __attribute__((ext_vector_type(8))) float v8f;
typedef __attribute__((ext_vector_type(4))) float f4;
typedef __attribute__((ext_vector_type(4))) unsigned int u32x4;
typedef __attribute__((ext_vector_type(4))) int i32x4;
typedef __attribute__((ext_vector_type(8))) int i32x8;

// ---------------------------------------------------------------------------
// TDM: DMA a tileH x 64-float 2D tile (row stride = ldN floats) from global
// memory into LDS at byte offset lds_byte. Tracked by TENSORcnt.
// ---------------------------------------------------------------------------
#if __has_builtin(__builtin_amdgcn_tensor_load_to_lds)
#define HAVE_TDM 1
__device__ __forceinline__ void tdm_load_tile(const float* gptr,
                                              unsigned int lds_byte,
                                              int ldN, int tileH) {
    unsigned long long ga = (unsigned long long)(uintptr_t)gptr;
    u32x4 g0;
    g0.x = 1u;                                        // count=1, no gather
    g0.y = lds_byte;                                  // LDS byte address
    g0.z = (unsigned int)(ga & 0xffffffffu);          // global_addr[31:0]
    g0.w = (unsigned int)((ga >> 32) & 0x1ffffffu)    // global_addr[56:32]
         | (2u << 30);                                // type = 2 (image)
    i32x8 g1 = (i32x8)0;
    g1[0] = (int)(2u << 16);                          // data_size = 4 bytes
    g1[1] = (int)((64u & 0xffffu) << 16);             // tensor_dim0 = 64 (lo16)
    g1[2] = (int)(((unsigned)tileH & 0xffffu) << 16); // dim0 hi=0 | tensor_dim1 lo16
    g1[3] = (int)(64u << 16);                         // dim1 hi=0 | tile_dim0 = 64
    g1[4] = (int)((unsigned)tileH & 0xffffu);         // tile_dim1 | tile_dim2=0
    g1[5] = ldN;                                      // tensor_dim0_stride[31:0]
    i32x4 z4 = (i32x4)0;
#if defined(__clang_major__) && (__clang_major__ >= 23)
    i32x8 z8 = (i32x8)0;
    __builtin_amdgcn_tensor_load_to_lds(g0, g1, z4, z4, z8, 0);
#else
    __builtin_amdgcn_tensor_load_to_lds(g0, g1, z4, z4, 0);
#endif
}
#else
#define HAVE_TDM 0
#endif

// ---------------------------------------------------------------------------
// 1) Embedding gather
// ---------------------------------------------------------------------------
__global__ void embed_gather_kernel(const int* __restrict__ feats,
                                    const float* __restrict__ embedding,
                                    float* __restrict__ x0) {
    int tid = blockIdx.x * blockDim.x + threadIdx.x;   // over NNZ*4
    if (tid >= NNZ * 4) return;
    int fg = tid >> 2;
    int q  = tid & 3;
    int feat = feats[fg];
    const f4* src = (const f4*)(embedding + (size_t)feat * K_EMB) + q;
    f4* dst = (f4*)(x0 + (size_t)fg * K_EMB) + q;
    *dst = *src;
}

// ---------------------------------------------------------------------------
// 2) First order
// ---------------------------------------------------------------------------
__global__ void first_order_kernel(const int* __restrict__ feats,
                                   const float* __restrict__ values,
                                   const float* __restrict__ weight_tab,
                                   const float* __restrict__ bias,
                                   float* __restrict__ first) {
    int b = blockIdx.x * blockDim.x + threadIdx.x;
    if (b >= B_SZ) return;
    float s = 0.f;
    int base = b * N_FIELDS;
    #pragma unroll 13
    for (int f = 0; f < N_FIELDS; ++f) {
        s += weight_tab[feats[base + f]] * values[base + f];
    }
    first[b] = s + bias[0];
}

// ---------------------------------------------------------------------------
// 3) Cross network: one block (128 thr) per sample
// ---------------------------------------------------------------------------
__global__ void cross_kernel(const float* __restrict__ x0,
                             const float* __restrict__ wc,
                             const float* __restrict__ bc,
                             float* __restrict__ cross) {
    const int b   = blockIdx.x;
    const int tid = threadIdx.x;            // 0..127
    const float* xrow = x0 + (size_t)b * XDIM;

    float x0r[5], xlr[5];
    #pragma unroll
    for (int i = 0; i < 5; ++i) {
        int idx = tid + i * 128;
        float v = (idx < XDIM) ? xrow[idx] : 0.f;
        x0r[i] = v;
        xlr[i] = v;
    }

    __shared__ float red[128];
    __shared__ float tbc;

    for (int d = 0; d < 3; ++d) {
        float p = 0.f;
        #pragma unroll
        for (int i = 0; i < 5; ++i) {
            int idx = tid + i * 128;
            if (idx < XDIM) p += xlr[i] * wc[d * XDIM + idx];
        }
        red[tid] = p;
        __syncthreads();
        for (int s = 64; s > 0; s >>= 1) {
            if (tid < s) red[tid] += red[tid + s];
            __syncthreads();
        }
        if (tid == 0) tbc = red[0];
        __syncthreads();
        float t = tbc;
        float bcd = bc[d];
        #pragma unroll
        for (int i = 0; i < 5; ++i) {
            xlr[i] = x0r[i] * t + bcd + xlr[i];
        }
        __syncthreads();
    }

    float* crow = cross + (size_t)b * XDIM;
    #pragma unroll
    for (int i = 0; i < 5; ++i) {
        int idx = tid + i * 128;
        if (idx < XDIM) crow[idx] = xlr[i];
    }
}

// ---------------------------------------------------------------------------
// 16-K sub-block: preload 8 A fragments (global b64) + 16 B fragments (LDS),
// then 32 back-to-back V_WMMA_F32_16X16X4_F32. All indices compile-time.
// Fragment layouts per ISA 7.12.2.
// ---------------------------------------------------------------------------
__device__ __forceinline__ void gemm_sub16(const float* __restrict__ bp,
                                           int kkBase,
                                           const float* __restrict__ a0p,
                                           const float* __restrict__ a1p,
                                           int half2, int lr,
                                           v8f (&acc)[2][4]) {
    v2f a0[4], a1[4];
    #pragma unroll
    for (int s = 0; s < 4; ++s) {
        a0[s] = *(const v2f*)(a0p + 4 * s + half2);
        a1[s] = *(const v2f*)(a1p + 4 * s + half2);
    }
    v2f bfr[4][4];
    #pragma unroll
    for (int s = 0; s < 4; ++s) {
        const int kk = kkBase + 4 * s + half2;
        #pragma unroll
        for (int j = 0; j < 4; ++j) {
            const int n = (j << 4) + lr;
            bfr[s][j].x = bp[kk * 64 + n];
            bfr[s][j].y = bp[(kk + 1) * 64 + n];
        }
    }
    #pragma unroll
    for (int s = 0; s < 4; ++s) {
        #pragma unroll
        for (int j = 0; j < 4; ++j) {
            acc[0][j] = __builtin_amdgcn_wmma_f32_16x16x4_f32(
                false, a0[s], false, bfr[s][j], (short)0, acc[0][j], false, false);
            acc[1][j] = __builtin_amdgcn_wmma_f32_16x16x4_f32(
                false, a1[s], false, bfr[s][j], (short)0, acc[1][j], false, false);
        }
    }
}

// ---------------------------------------------------------------------------
// 4) fp32 WMMA GEMM + bias + ReLU:  C[M,N] = relu(A[M,K] @ W[K,N] + b)
// Block: 256 threads = 8 waves; block tile 256(M) x 64(N); per-wave 32x64.
// B panel (KC x 64, 16 KB) TDM-streamed into LDS, double-buffered.
// Templated on N/K so chunk counts and the 624-tail are compile-time.
// ---------------------------------------------------------------------------
template<int N, int K, int RELU>
__global__ __launch_bounds__(256) void wmma_gemm_kernel(
        const float* __restrict__ A,
        const float* __restrict__ W,
        const float* __restrict__ bias,
        float* __restrict__ C) {
    __shared__ float ldsW[2][KC * 64];     // 2 x 16 KB double buffer

    constexpr int FULL = K / KC;           // full 64-K chunks
    constexpr int TAIL = K % KC;           // 0 or 48 (multiple of 16)
    constexpr int NCH  = FULL + (TAIL ? 1 : 0);

    const int lane = threadIdx.x & 31;
    const int wave = threadIdx.x >> 5;
    const int mBlocks = B_SZ >> 8;         // 256-row block tiles
    const int bm = blockIdx.x % mBlocks;
    const int bn = blockIdx.x / mBlocks;
    const int m0 = (bm << 8) + (wave << 5);
    const int n0 = bn << 6;

    const int half2 = (lane >> 4) << 1;    // 0 or 2
    const int lr    = lane & 15;

    v8f acc[2][4];
    #pragma unroll
    for (int i = 0; i < 2; ++i)
        #pragma unroll
        for (int j = 0; j < 4; ++j)
            acc[i][j] = (v8f)0.f;

    const float* Arow0 = A + (size_t)(m0 + lr) * K;
    const float* Arow1 = A + (size_t)(m0 + 16 + lr) * K;

#if HAVE_TDM
    if (wave == 0) {
        tdm_load_tile(W + n0, (unsigned int)(uintptr_t)&ldsW[0][0], N,
                      FULL ? KC : TAIL);
    }
#endif

    int buf = 0;
    #pragma unroll 1
    for (int c = 0; c < FULL; ++c) {
        const int k0 = c * KC;

#if HAVE_TDM
        if (wave == 0) __builtin_amdgcn_s_wait_tensorcnt(0);  // chunk c landed
        __syncthreads();
        if (c + 1 < NCH && wave == 0) {
            const int k1 = k0 + KC;
            tdm_load_tile(W + (size_t)k1 * N + n0,
                          (unsigned int)(uintptr_t)&ldsW[buf ^ 1][0], N,
                          (c + 1 < FULL) ? KC : TAIL);
        }
#else
        for (int q = threadIdx.x; q < KC * 16; q += 256) {
            int kk = q >> 4;
            int n4 = q & 15;
            *(f4*)&ldsW[buf][kk * 64 + n4 * 4] =
                *(const f4*)(W + (size_t)(k0 + kk) * N + n0 + n4 * 4);
        }
        __syncthreads();
#endif
        // prefetch next A panel (gfx1250 global_prefetch_b8)
        __builtin_prefetch(Arow0 + k0 + KC, 0, 1);

        const float* bp = &ldsW[buf][0];
        #pragma unroll
        for (int sb = 0; sb < 4; ++sb) {       // 4 x 16-K sub-blocks
            gemm_sub16(bp, sb * 16, Arow0 + k0 + sb * 16, Arow1 + k0 + sb * 16,
                       half2, lr, acc);
        }
        __syncthreads();                       // buf free for chunk c+2 DMA
        buf ^= 1;
    }

    if (TAIL > 0) {                            // constexpr-folded epilogue
        const int k0 = FULL * KC;
#if HAVE_TDM
        if (wave == 0) __builtin_amdgcn_s_wait_tensorcnt(0);
        __syncthreads();
#else
        for (int q = threadIdx.x; q < TAIL * 16; q += 256) {
            int kk = q >> 4;
            int n4 = q & 15;
            *(f4*)&ldsW[buf][kk * 64 + n4 * 4] =
                *(const f4*)(W + (size_t)(k0 + kk) * N + n0 + n4 * 4);
        }
        __syncthreads();
#endif
        const float* bp = &ldsW[buf][0];
        #pragma unroll
        for (int sb = 0; sb < TAIL / 16; ++sb) {
            gemm_sub16(bp, sb * 16, Arow0 + k0 + sb * 16, Arow1 + k0 + sb * 16,
                       half2, lr, acc);
        }
        __syncthreads();
    }

    const float bb = bias[0];
    #pragma unroll
    for (int i = 0; i < 2; ++i) {
        #pragma unroll
        for (int j = 0; j < 4; ++j) {
            const int n = n0 + (j << 4) + lr;
            #pragma unroll
            for (int v = 0; v < 8; ++v) {
                const int m = m0 + (i << 4) + v + ((lane >> 4) << 3);
                float val = acc[i][j][v] + bb;
                if (RELU) val = fmaxf(val, 0.f);
                C[(size_t)m * N + n] = val;
            }
        }
    }
}

// ---------------------------------------------------------------------------
// 5) Output: row i gathers concat(cross.flat, h.flat)[i*880 .. +880)
// ---------------------------------------------------------------------------
__global__ void output_kernel(const float* __restrict__ cross,
                              const float* __restrict__ h3,
                              const float* __restrict__ w_out,
                              const float* __restrict__ first,
                              const float* __restrict__ b_out,
                              float* __restrict__ out) {
    const int b   = blockIdx.x;
    const int tid = threadIdx.x;           // 0..255
    const size_t base  = (size_t)b * ZDIM;
    const size_t split = (size_t)B_SZ * XDIM;

    float p = 0.f;
    for (int j = tid; j < ZDIM; j += 256) {
        size_t g = base + (size_t)j;
        float v = (g < split) ? cross[g] : h3[g - split];
        p += v * w_out[j];
    }

    __shared__ float red[256];
    red[tid] = p;
    __syncthreads();
    for (int s = 128; s > 0; s >>= 1) {
        if (tid < s) red[tid] += red[tid + s];
        __syncthreads();
    }
    if (tid == 0) {
        float z = red[0] + first[b] + b_out[0];
        out[b] = 1.f / (1.f + expf(-z));
    }
}

// ---------------------------------------------------------------------------
// Launch
// ---------------------------------------------------------------------------
extern "C" void kernel_launch(void* const* d_in, const int* in_sizes, int n_in,
                              void* d_out, int out_size, void* d_ws, size_t ws_size,
                              hipStream_t stream) {
    // 0 batch_size, 1 index, 2 feats, 3 values, 4 bias, 5 weight_tab,
    // 6 embedding, 7 wc, 8 bc, 9 W1, 10 b1, 11 W2, 12 b2, 13 W3, 14 b3,
    // 15 w_out, 16 b_out
    const int*   feats      = (const int*)  d_in[2];
    const float* values     = (const float*)d_in[3];
    const float* bias       = (const float*)d_in[4];
    const float* weight_tab = (const float*)d_in[5];
    const float* embedding  = (const float*)d_in[6];
    const float* wc         = (const float*)d_in[7];
    const float* bc         = (const float*)d_in[8];
    const float* W1         = (const float*)d_in[9];
    const float* b1         = (const float*)d_in[10];
    const float* W2         = (const float*)d_in[11];
    const float* b2         = (const float*)d_in[12];
    const float* W3         = (const float*)d_in[13];
    const float* b3         = (const float*)d_in[14];
    const float* w_out      = (const float*)d_in[15];
    const float* b_out      = (const float*)d_in[16];
    float* out = (float*)d_out;

    // Workspace: first | x0 | cross | h1 ; h2 aliases x0, h3 aliases h1.
    float* ws    = (float*)d_ws;
    float* first = ws;                                   // B
    float* x0    = first + B_SZ;                         // B*624
    float* cross = x0    + (size_t)B_SZ * XDIM;          // B*624
    float* h1    = cross + (size_t)B_SZ * XDIM;          // B*1024
    float* h2    = x0;                                   // alias: B*512
    float* h3    = h1;                                   // alias: B*256

    {
        int n = NNZ * 4;
        embed_gather_kernel<<<(n + 255) / 256, 256, 0, stream>>>(feats, embedding, x0);
    }
    first_order_kernel<<<(B_SZ + 255) / 256, 256, 0, stream>>>(
        feats, values, weight_tab, bias, first);
    cross_kernel<<<B_SZ, 128, 0, stream>>>(x0, wc, bc, cross);

    wmma_gemm_kernel<FC1, XDIM, 1><<<(B_SZ / 256) * (FC1 / 64), 256, 0, stream>>>(
        x0, W1, b1, h1);
    wmma_gemm_kernel<FC2, FC1, 1><<<(B_SZ / 256) * (FC2 / 64), 256, 0, stream>>>(
        h1, W2, b2, h2);
    wmma_gemm_kernel<FC3, FC2, 1><<<(B_SZ / 256) * (FC3 / 64), 256, 0, stream>>>(
        h2, W3, b3, h3);

    output_kernel<<<B_SZ, 256, 0, stream>>>(cross, h3, w_out, first, b_out, out);
}